// NeighborhoodGCN_52055003627816
// MI455X (gfx1250) — compile-verified
//
#include <hip/hip_runtime.h>
#include <hip/hip_bf16.h>

typedef __attribute__((ext_vector_type(2))) float v2f;
typedef __attribute__((ext_vector_type(8))) float v8f;

// ---------------------------------------------------------------------------
// 1) deg[i] = 1.0 (self loop weight)
__global__ void k_init_deg(float* deg, int n) {
    int i = blockIdx.x * blockDim.x + threadIdx.x;
    if (i < n) deg[i] = 1.0f;
}

// 2) deg[dst[e]] += ew[e]
__global__ void k_deg_accum(const int* ei, const float* ew, float* deg, int E) {
    int e = blockIdx.x * blockDim.x + threadIdx.x;
    if (e < E) {
        int d = ei[E + e];
        atomicAdd(&deg[d], ew[e]);
    }
}

// 3) dinv[i] = deg>0 ? rsqrt(deg) : 0
__global__ void k_dinv(const float* deg, float* dinv, int n) {
    int i = blockIdx.x * blockDim.x + threadIdx.x;
    if (i < n) {
        float d = deg[i];
        dinv[i] = (d > 0.0f) ? rsqrtf(d) : 0.0f;
    }
}

// 4) xw[N,16] = x[N,2] @ W1[2,16] via V_WMMA_F32_16X16X4_F32.
//    One wave32 per 16-row tile.  K=2 real, K=2..3 zero padded.
//    A layout (32-bit 16x4): lanes 0-15 hold K=0,1 ; lanes 16-31 hold K=2,3 (zeros).
//    B layout (32-bit 4x16): lanes 0-15 col N=lane, VGPR0=K0,VGPR1=K1; lanes 16-31 K=2,3 (zeros).
//    D layout: VGPR j, lanes0-15 -> M=j, lanes16-31 -> M=j+8 ; N = lane%16.
__global__ void k_xw_wmma(const float* __restrict__ x, const float* __restrict__ W1,
                          float* __restrict__ xw, int n, int ntiles) {
    int wave = threadIdx.x >> 5;
    int lane = threadIdx.x & 31;
    int tile = blockIdx.x * (blockDim.x >> 5) + wave;
    if (tile >= ntiles) return;           // wave-uniform exit, EXEC stays full inside

    int tileBase = tile * 16;
    int half = lane >> 4;                 // 0: lanes 0-15, 1: lanes 16-31
    int l16  = lane & 15;
    bool fullTile = (tileBase + 16 <= n); // wave-uniform

    v2f a = {0.0f, 0.0f};
    v2f b = {0.0f, 0.0f};
    if (half == 0) {
        int row = tileBase + l16;
        if (row < n) {
            a.x = x[row * 2 + 0];         // K=0
            a.y = x[row * 2 + 1];         // K=1
        }
        b.x = W1[0 * 16 + l16];           // K=0 row of W1
        b.y = W1[1 * 16 + l16];           // K=1 row of W1
    }
    v8f c = {};
    v8f d = __builtin_amdgcn_wmma_f32_16x16x4_f32(
        /*neg_a=*/false, a, /*neg_b=*/false, b,
        /*c_mod=*/(short)0, c, /*reuse_a=*/false, /*reuse_b=*/false);

    // D element j of lane: row = tileBase + j + half*8, col = l16.
    float* base = xw + (size_t)(tileBase + half * 8) * 16 + l16;
    if (fullTile) {
        // unguarded: 8 stores, each half-wave writes 64B contiguous
#pragma unroll
        for (int j = 0; j < 8; ++j) base[j * 16] = d[j];
    } else {
#pragma unroll
        for (int j = 0; j < 8; ++j) {
            int row = tileBase + half * 8 + j;
            if (row < n) base[j * 16] = d[j];
        }
    }
}

// 5) norm[e] = dinv[src]*ew*dinv[dst]   (reused by both layers)
__global__ void k_norm(const int* ei, const float* ew, const float* dinv,
                       float* norm, int E) {
    int e = blockIdx.x * blockDim.x + threadIdx.x;
    if (e < E) {
        int s = ei[e];
        int d = ei[E + e];
        norm[e] = dinv[s] * ew[e] * dinv[d];
    }
}

// 6) h init with self-loop message: h[i,c] = xw[i,c] * dinv[i]^2
__global__ void k_h_init(const float* __restrict__ xw, const float* __restrict__ dinv,
                         float* __restrict__ h, int n) {
    int t = blockIdx.x * blockDim.x + threadIdx.x;
    if (t < n * 16) {
        int i = t >> 4;
        float di = dinv[i];
        h[t] = xw[t] * di * di;
    }
}

// 7) layer-1 edge scatter: 16 lanes per edge, channel = lane&15.
//    Gather xw[src*16+c] (64B contiguous per edge) and atomic-add into h[dst*16+c].
__global__ void k_edge1(const int* __restrict__ ei, const float* __restrict__ norm,
                        const float* __restrict__ xw, float* __restrict__ h,
                        int E) {
    long long t = (long long)blockIdx.x * blockDim.x + threadIdx.x;
    long long e = t >> 4;
    int c = (int)(t & 15);
    if (e < E) {
        int s = ei[e];
        int d = ei[E + e];
        float nm = norm[e];
        float msg = xw[s * 16 + c] * nm;
        atomicAdd(&h[d * 16 + c], msg);
    }
}

// 8) hw[i] = sum_c relu(h[i,c] + b1[c]) * W2[c]
__global__ void k_relu_dot(const float* __restrict__ h, const float* __restrict__ b1,
                           const float* __restrict__ W2, float* __restrict__ hw, int n) {
    int i = blockIdx.x * blockDim.x + threadIdx.x;
    if (i < n) {
        float acc = 0.0f;
#pragma unroll
        for (int c = 0; c < 16; ++c) {
            float v = h[i * 16 + c] + b1[c];
            v = v > 0.0f ? v : 0.0f;
            acc = fmaf(v, W2[c], acc);
        }
        hw[i] = acc;
    }
}

// 9) out init with self-loop message + bias: out[i] = hw[i]*dinv[i]^2 + b2
__global__ void k_out_init(const float* __restrict__ hw, const float* __restrict__ dinv,
                           const float* __restrict__ b2, float* __restrict__ out, int n) {
    int i = blockIdx.x * blockDim.x + threadIdx.x;
    if (i < n) {
        float di = dinv[i];
        out[i] = hw[i] * di * di + b2[0];
    }
}

// 10) layer-2 edge scatter: out[dst] += hw[src]*norm
__global__ void k_edge2(const int* __restrict__ ei, const float* __restrict__ norm,
                        const float* __restrict__ hw, float* __restrict__ out, int E) {
    int e = blockIdx.x * blockDim.x + threadIdx.x;
    if (e < E) {
        int s = ei[e];
        int d = ei[E + e];
        atomicAdd(&out[d], hw[s] * norm[e]);
    }
}

extern "C" void kernel_launch(void* const* d_in, const int* in_sizes, int n_in,
                              void* d_out, int out_size, void* d_ws, size_t ws_size,
                              hipStream_t stream) {
    const float* x   = (const float*)d_in[0];   // [N,2]
    const int*   ei  = (const int*)  d_in[1];   // [2,E]
    const float* ew  = (const float*)d_in[2];   // [E]
    const float* W1  = (const float*)d_in[3];   // [2,16]
    const float* b1  = (const float*)d_in[4];   // [16]
    const float* W2  = (const float*)d_in[5];   // [16,1]
    const float* b2  = (const float*)d_in[6];   // [1]
    float* out = (float*)d_out;                 // [N]

    const int N = in_sizes[0] / 2;
    const int E = in_sizes[2];

    // workspace layout (floats)
    float* ws   = (float*)d_ws;
    float* deg  = ws;                    // N
    float* dinv = ws + (size_t)N;        // N
    float* hw   = ws + (size_t)2 * N;    // N
    float* xw   = ws + (size_t)3 * N;    // 16N
    float* h    = ws + (size_t)19 * N;   // 16N
    float* norm = ws + (size_t)35 * N;   // E

    const int B = 256;
    auto gb = [](long long work, int blk) { return (int)((work + blk - 1) / blk); };

    k_init_deg <<<gb(N, B), B, 0, stream>>>(deg, N);
    k_deg_accum<<<gb(E, B), B, 0, stream>>>(ei, ew, deg, E);
    k_dinv     <<<gb(N, B), B, 0, stream>>>(deg, dinv, N);

    int ntiles = (N + 15) / 16;
    k_xw_wmma  <<<gb((long long)ntiles * 32, B), B, 0, stream>>>(x, W1, xw, N, ntiles);

    k_norm     <<<gb(E, B), B, 0, stream>>>(ei, ew, dinv, norm, E);
    k_h_init   <<<gb((long long)N * 16, B), B, 0, stream>>>(xw, dinv, h, N);
    k_edge1    <<<gb((long long)E * 16, B), B, 0, stream>>>(ei, norm, xw, h, E);
    k_relu_dot <<<gb(N, B), B, 0, stream>>>(h, b1, W2, hw, N);
    k_out_init <<<gb(N, B), B, 0, stream>>>(hw, dinv, b2, out, N);
    k_edge2    <<<gb(E, B), B, 0, stream>>>(ei, norm, hw, out, E);
}